// QuantizedSLNRPrecoder_85950885527899
// MI455X (gfx1250) — compile-verified
//
#include <hip/hip_runtime.h>

// MI455X / gfx1250, wave32. One wave solves one (t,f) SLNR precoding problem:
//   R = H^H H  (16x16x8 complex Gram)  -> 8x v_wmma_f32_16x16x4_f32
//   W = (R + sigma2*I)^-1 H^H          -> wave32 column-parallel Gauss-Jordan
//       (pivot column broadcast via v_readlane -> SGPR scalar operands)
//   normalize columns; G and x_precoded stored coalesced via LDS transpose.

typedef __attribute__((ext_vector_type(2))) float v2f;
typedef __attribute__((ext_vector_type(8))) float v8f;

#define TDIM   14
#define FDIM   4096
#define SDIM   8          // gathered streams (NUM_UE*ANT_PER_UE == S)
#define NTANT  16
#define FPB    8          // f-points per block (one per wave)
#define XP_CPLX ((size_t)NTANT * TDIM * FDIM)   // x_precoded complex elements

__device__ __forceinline__ float rl(float v, int l) {
    return __int_as_float(__builtin_amdgcn_readlane(__float_as_int(v), l));
}

__global__ __launch_bounds__(256)
void slnr_precoder_kernel(const float* __restrict__ xr, const float* __restrict__ xi,
                          const float* __restrict__ hr, const float* __restrict__ hi,
                          const float* __restrict__ snr_db,
                          const int*  __restrict__ sched,
                          float* __restrict__ out)
{
    __shared__ float  sHr[FPB][SDIM][NTANT];
    __shared__ float  sHi[FPB][SDIM][NTANT];
    __shared__ float  sXr[FPB][SDIM];
    __shared__ float  sXi[FPB][SDIM];
    __shared__ float2 sG[FPB][NTANT][SDIM];   // W transpose buffer (8 KB)
    __shared__ float2 sY[FPB][NTANT];         // y staging (1 KB)

    const int tid = threadIdx.x;
    const int t   = blockIdx.x / (FDIM / FPB);
    const int f0  = (blockIdx.x % (FDIM / FPB)) * FPB;

    // ---- cooperative staging: global -> LDS (8 consecutive f per (s,n) pair) ----
    #pragma unroll
    for (int i = 0; i < 4; ++i) {
        int e    = tid + i * 256;      // 0..1023
        int prob = e & 7;
        int pair = e >> 3;             // 0..127
        int s    = pair >> 4;
        int n    = pair & 15;
        int src  = sched[s];           // gather along S axis
        size_t g = ((size_t)(src * NTANT + n) * TDIM + t) * FDIM + (size_t)(f0 + prob);
        sHr[prob][s][n] = hr[g];
        sHi[prob][s][n] = hi[g];
    }
    if (tid < 64) {
        int prob = tid & 7, s = tid >> 3;
        size_t g = ((size_t)s * TDIM + t) * FDIM + (size_t)(f0 + prob);
        sXr[prob][s] = xr[g];
        sXi[prob][s] = xi[g];
    }
    __syncthreads();

    const int wave = tid >> 5;
    const int lane = tid & 31;
    const int f    = f0 + wave;
    const float sigma2 = __powf(10.0f, -0.1f * snr_db[0]);

    // ---- WMMA operands: A = H^T slice (16x4 f32 layout), B identical for Gram ----
    const int m  = lane & 15;
    const int kh = lane >> 4;          // K-half select per A/B f32 layout
    v2f ar0, ar1, ai0, ai1;
    ar0.x = sHr[wave][2*kh + 0][m];  ar0.y = sHr[wave][2*kh + 1][m];
    ar1.x = sHr[wave][2*kh + 4][m];  ar1.y = sHr[wave][2*kh + 5][m];
    ai0.x = sHi[wave][2*kh + 0][m];  ai0.y = sHi[wave][2*kh + 1][m];
    ai1.x = sHi[wave][2*kh + 4][m];  ai1.y = sHi[wave][2*kh + 5][m];

    // ---- R = H^H H via 8 chained f32 WMMAs (K = 8 -> 2x K=4 per product) ----
    v8f accz = {};
    v8f rre, rim;
    rre = __builtin_amdgcn_wmma_f32_16x16x4_f32(false, ai1, false, ai1, (short)0, accz, false, false);
    rre = __builtin_amdgcn_wmma_f32_16x16x4_f32(false, ai0, false, ai0, (short)0, rre,  false, false);
    rre = __builtin_amdgcn_wmma_f32_16x16x4_f32(false, ar1, false, ar1, (short)0, rre,  false, false);
    rre = __builtin_amdgcn_wmma_f32_16x16x4_f32(false, ar0, false, ar0, (short)0, rre,  false, false);
    v2f nai0 = -ai0, nai1 = -ai1;     // fold the minus of -Hi^T Hr into A
    rim = __builtin_amdgcn_wmma_f32_16x16x4_f32(false, nai1, false, ar1, (short)0, accz, false, false);
    rim = __builtin_amdgcn_wmma_f32_16x16x4_f32(false, nai0, false, ar0, (short)0, rim,  false, false);
    rim = __builtin_amdgcn_wmma_f32_16x16x4_f32(false, ar1,  false, ai1, (short)0, rim,  false, false);
    rim = __builtin_amdgcn_wmma_f32_16x16x4_f32(false, ar0,  false, ai0, (short)0, rim,  false, false);

    // ---- gather full column c of R into lane c (rows 8..15 live in lane c+16) ----
    float cr[16], ci[16];
    #pragma unroll
    for (int v = 0; v < 8; ++v) {
        float hiR = __shfl_xor(rre[v], 16, 32);
        float hiI = __shfl_xor(rim[v], 16, 32);
        cr[v]     = rre[v];  ci[v]     = rim[v];
        cr[v + 8] = hiR;     ci[v + 8] = hiI;
    }
    // M = R + sigma2*I  (diagonal entry of column 'lane' is row 'lane')
    #pragma unroll
    for (int r = 0; r < 16; ++r)
        cr[r] += (lane == r) ? sigma2 : 0.0f;
    // lanes 16..23: RHS columns  Hh[:,s] = conj(H[s,:]); lanes 24..31 mirror (inert)
    if (lane >= 16) {
        int s = (lane - 16) & 7;
        #pragma unroll
        for (int r = 0; r < 16; ++r) {
            cr[r] =  sHr[wave][s][r];
            ci[r] = -sHi[wave][s][r];
        }
    }

    // ---- Gauss-Jordan on [M | Hh], one column per lane (24 live columns) ----
    // Pivot-column values come from lane k via v_readlane (uniform SGPR broadcast).
    #pragma unroll
    for (int k = 0; k < 16; ++k) {
        float pr  = rl(cr[k], k);
        float pi  = rl(ci[k], k);
        float inv = 1.0f / (pr * pr + pi * pi);   // scalar (SALU/VALU-scalar) math
        float qr  =  pr * inv;
        float qi  = -pi * inv;
        float akr = cr[k] * qr - ci[k] * qi;      // row k scaled by 1/pivot
        float aki = cr[k] * qi + ci[k] * qr;
        cr[k] = akr; ci[k] = aki;
        #pragma unroll
        for (int r = 0; r < 16; ++r) {
            if (r == k) continue;
            float mr = rl(cr[r], k);              // multiplier A[r][k] (SGPR)
            float mi = rl(ci[r], k);
            cr[r] = fmaf(-mr, akr, fmaf( mi, aki, cr[r]));
            ci[r] = fmaf(-mr, aki, fmaf(-mi, akr, ci[r]));
        }
    }

    // ---- column 2-norm normalization (lane-local; M-columns are unit vectors) ----
    float n2 = 0.0f;
    #pragma unroll
    for (int r = 0; r < 16; ++r) n2 += cr[r] * cr[r] + ci[r] * ci[r];
    float scl = rsqrtf(n2);
    #pragma unroll
    for (int r = 0; r < 16; ++r) { cr[r] *= scl; ci[r] *= scl; }

    // ---- G: LDS transpose then fully-coalesced b128 stores (block region is
    //      contiguous: 8 probs x 16 n x 8 s complex = 8 KB) ----
    if (lane >= 16 && lane < 24) {
        int s = lane - 16;
        #pragma unroll
        for (int n = 0; n < 16; ++n)
            sG[wave][n][s] = make_float2(cr[n], ci[n]);
    }
    __syncthreads();
    {
        const float4* sGv = (const float4*)&sG[0][0][0];           // 512 float4
        float4* gOut = (float4*)(out + 2 * XP_CPLX
                                 + ((size_t)t * FDIM + f0) * NTANT * SDIM * 2);
        gOut[tid]       = sGv[tid];
        gOut[tid + 256] = sGv[tid + 256];
    }

    // ---- y[n] = sum_s W[n][s] * x[s]: per-lane partials + 8-lane butterfly ----
    int sx = (lane - 16) & 7;
    float xre = sXr[wave][sx];
    float xim = sXi[wave][sx];
    float yr[16], yi[16];
    #pragma unroll
    for (int n = 0; n < 16; ++n) {
        yr[n] = cr[n] * xre - ci[n] * xim;
        yi[n] = cr[n] * xim + ci[n] * xre;
    }
    #pragma unroll
    for (int n = 0; n < 16; ++n) {
        yr[n] += __shfl_xor(yr[n], 1, 32);  yi[n] += __shfl_xor(yi[n], 1, 32);
        yr[n] += __shfl_xor(yr[n], 2, 32);  yi[n] += __shfl_xor(yi[n], 2, 32);
        yr[n] += __shfl_xor(yr[n], 4, 32);  yi[n] += __shfl_xor(yi[n], 4, 32);
    }
    if (lane == 16) {
        #pragma unroll
        for (int n = 0; n < 16; ++n)
            sY[wave][n] = make_float2(yr[n], yi[n]);
    }
    __syncthreads();
    if (tid < 128) {   // 64 B contiguous runs per n across the block's 8 f-points
        int n = tid >> 3, prob = tid & 7;
        float2* xpo = (float2*)out;    // x_precoded (b,u,n,t,f)
        xpo[((size_t)n * TDIM + t) * FDIM + f0 + prob] = sY[prob][n];
    }
}

extern "C" void kernel_launch(void* const* d_in, const int* in_sizes, int n_in,
                              void* d_out, int out_size, void* d_ws, size_t ws_size,
                              hipStream_t stream) {
    (void)in_sizes; (void)n_in; (void)out_size; (void)d_ws; (void)ws_size;
    const float* xr   = (const float*)d_in[0];
    const float* xi   = (const float*)d_in[1];
    const float* hr   = (const float*)d_in[2];
    const float* hi   = (const float*)d_in[3];
    const float* snr  = (const float*)d_in[4];
    const int*   schd = (const int*)d_in[5];
    // d_in[6] = ue_ranks: unused by the reference
    float* out = (float*)d_out;

    dim3 grid(TDIM * (FDIM / FPB));   // 14 * 512 = 7168 blocks
    dim3 block(256);                  // 8 waves, one (t,f) problem each
    slnr_precoder_kernel<<<grid, block, 0, stream>>>(xr, xi, hr, hi, snr, schd, out);
}